// ComponentToPair_55379308314761
// MI455X (gfx1250) — compile-verified
//
#include <hip/hip_runtime.h>
#include <hip/hip_bf16.h>

// ---------------------------------------------------------------------------
// out[b,i,j,f] = pi[b,i,f] + pj[b,j,f] + bias[f]
//   pi = X · W[:, :256]^T   (einsum 'bse,fe->bsf')
//   pj = X · W[:, 256:]^T
// B=4, S=256, E=256.  Output = 256 MB -> bandwidth bound (~11.5us @ 23.3TB/s).
// Pass 1: f32 WMMA GEMM producing P[1024, 512] = [pi | pj] into d_ws (2 MB).
// Pass 2: streaming broadcast-add with non-temporal float4 stores.
// ---------------------------------------------------------------------------

typedef __attribute__((ext_vector_type(2))) float v2f;
typedef __attribute__((ext_vector_type(4))) float v4f;
typedef __attribute__((ext_vector_type(8))) float v8f;

#define E_DIM   256
#define M_TOT   1024   // B*S
#define N_TOT   512    // 2*E
#define W_LD    512    // row stride of W

// One wave computes one 16x16 tile of P.  2048 tiles -> 256 blocks x 8 waves.
__global__ void __launch_bounds__(256)
pairproj_gemm_wmma(const float* __restrict__ X,   // [1024, 256]
                   const float* __restrict__ W,   // [256, 512]
                   float*       __restrict__ P)   // [1024, 512]
{
    const int wave = (blockIdx.x * blockDim.x + threadIdx.x) >> 5;
    const int lane = threadIdx.x & 31;

    const int mt = wave & 63;          // 64 tiles along M (1024/16)
    const int nt = wave >> 6;          // 32 tiles along N (512/16)
    const int m0 = mt << 4;
    const int n0 = nt << 4;

    // Column n of P in [0,512): half = n/256 selects W1 vs W2 block,
    // fL = n%256 is the W row index (f).
    const int half = n0 >> 8;          // 0 or 1
    const int fL0  = n0 & 255;

    const int lm = lane & 15;          // 0..15
    const int hi = lane >> 4;          // 0 or 1

    // f32 WMMA fragment layout: K = hi*2 + vgpr for both A and B.
    // A[m, k]  : row m0+lm of X, float2 at column k + hi*2
    // B[k, n]  = W[fL0+n][half*256 + k] : row fL0+lm of W, float2 at col k+hi*2
    const float* arow = X + (size_t)(m0 + lm) * E_DIM + (hi << 1);
    const float* brow = W + (size_t)(fL0 + lm) * W_LD + (half << 8) + (hi << 1);

    v8f c = {0.f, 0.f, 0.f, 0.f, 0.f, 0.f, 0.f, 0.f};

#pragma unroll 8
    for (int k = 0; k < E_DIM; k += 4) {
        v2f a = *(const v2f*)(arow + k);
        v2f b = *(const v2f*)(brow + k);
        // 8 args: (neg_a, A, neg_b, B, c_mod, C, reuse_a, reuse_b)
        c = __builtin_amdgcn_wmma_f32_16x16x4_f32(
                false, a, false, b, (short)0, c, false, false);
    }

    // C/D layout: VGPR r, lane l -> (M = (l>>4)*8 + r, N = l&15)
    float* prow = P + (size_t)(m0 + (hi << 3)) * N_TOT + n0 + lm;
#pragma unroll
    for (int r = 0; r < 8; ++r)
        prow[(size_t)r * N_TOT] = c[r];
}

// Each block: one (b,i) pair, 4 consecutive j, all 256 f as float4.
// Grid = 1024 * 64 = 65536 blocks, 256 threads each.
__global__ void __launch_bounds__(256)
pairproj_broadcast(const float* __restrict__ P,     // [1024, 512] = [pi | pj]
                   const float* __restrict__ bias,  // [256]
                   float*       __restrict__ out)   // [4,256,256,256]
{
    const int bi = blockIdx.x >> 6;          // b*256 + i  (0..1023)
    const int j0 = (blockIdx.x & 63) << 2;   // 0..252 step 4
    const int b  = bi >> 8;

    const int g  = threadIdx.x >> 6;         // 0..3  -> j offset
    const int f4 = threadIdx.x & 63;         // float4 index over f
    const int j  = j0 + g;

    const v4f* pi = (const v4f*)(P + (size_t)bi * N_TOT);                    // pi[b,i,:]
    const v4f* pj = (const v4f*)(P + (size_t)((b << 8) + j) * N_TOT + 256);  // pj[b,j,:]
    const v4f* bs = (const v4f*)bias;

    v4f vi = pi[f4];   // L2-resident, reused across all j
    v4f vj = pj[f4];   // L2-resident, reused across all i
    v4f vb = bs[f4];

    v4f r;
    r.x = vi.x + vj.x + vb.x;
    r.y = vi.y + vj.y + vb.y;
    r.z = vi.z + vj.z + vb.z;
    r.w = vi.w + vj.w + vb.w;

    v4f* o = (v4f*)(out + ((size_t)bi * 256 + j) * E_DIM);
    __builtin_nontemporal_store(r, &o[f4]);   // 256MB stream: bypass caches
}

extern "C" void kernel_launch(void* const* d_in, const int* in_sizes, int n_in,
                              void* d_out, int out_size, void* d_ws, size_t ws_size,
                              hipStream_t stream) {
    const float* X    = (const float*)d_in[0];  // [4,256,256]  -> [1024,256]
    const float* W    = (const float*)d_in[1];  // [256,512]
    const float* bias = (const float*)d_in[2];  // [256]
    float*       out  = (float*)d_out;          // [4,256,256,256]
    float*       P    = (float*)d_ws;           // needs 1024*512*4 = 2 MB scratch

    // Pass 1: P = [pi | pj] via f32 WMMA. 2048 tiles = 256 blocks x 8 waves.
    pairproj_gemm_wmma<<<dim3(256), dim3(256), 0, stream>>>(X, W, P);

    // Pass 2: broadcast add, 65536 blocks x 256 threads, NT stores.
    pairproj_broadcast<<<dim3(65536), dim3(256), 0, stream>>>(P, bias, out);
}